// ReLU_13700945674664
// MI455X (gfx1250) — compile-verified
//
#include <hip/hip_runtime.h>
#include <stdint.h>

#ifndef __has_builtin
#define __has_builtin(x) 0
#endif

// CDNA5 async global->LDS staging path (gated: compiles with plain-LDS
// fallback if the toolchain lacks the builtins).
#if __has_builtin(__builtin_amdgcn_global_load_async_to_lds_b128) && \
    __has_builtin(__builtin_amdgcn_s_wait_asynccnt)
#define USE_ASYNC_LDS 1
#else
#define USE_ASYNC_LDS 0
#endif

typedef int v4i __attribute__((vector_size(16)));

// Builtin signature (from hipcc diagnostic): param0 = v4i in AS1 (global),
// param1 = v4i in AS3 (LDS), then imm offset, imm cpol. Int round-trips avoid
// illegal generic->AS pointer casts; generic LDS flat addr low 32 bits == LDS
// offset, generic global flat addr == AS1 addr.
#define AS1_V4I(p) \
  ((__attribute__((address_space(1))) v4i*)(uintptr_t)(p))
#define AS3_V4I(p) \
  ((__attribute__((address_space(3))) v4i*)(uint32_t)(uintptr_t)(p))

#define DD 1024          // weight columns per row
#define ROWLEN (DD + 1)  // + bias
#define EPL 32           // elements per lane (DD / 32 lanes)

__global__ __launch_bounds__(256) void relu_deeppoly_fused(
    const float* __restrict__ l, const float* __restrict__ u,
    const float* __restrict__ il, const float* __restrict__ iu,
    float* __restrict__ out_l, float* __restrict__ out_u,
    float* __restrict__ out_clb, float* __restrict__ out_cub, int rows) {
  __shared__ float s_il[DD];
  __shared__ float s_iu[DD];

  const int t = threadIdx.x;
  // ---- stage input_l / input_u (8 KB total) into LDS once per block ----
#if USE_ASYNC_LDS
  {
    const int i = t * 4;  // 256 threads * 4 floats == 1024
    __builtin_amdgcn_global_load_async_to_lds_b128(AS1_V4I(il + i),
                                                   AS3_V4I(s_il + i), 0, 0);
    __builtin_amdgcn_global_load_async_to_lds_b128(AS1_V4I(iu + i),
                                                   AS3_V4I(s_iu + i), 0, 0);
    __builtin_amdgcn_s_wait_asynccnt(0);
  }
  __syncthreads();
#else
  {
    const int i = t * 4;
    *(float4*)(s_il + i) = *(const float4*)(il + i);
    *(float4*)(s_iu + i) = *(const float4*)(iu + i);
  }
  __syncthreads();
#endif

  const int lane = t & 31;
  const int wid = (blockIdx.x * (blockDim.x >> 5)) + (t >> 5);
  const int nw = gridDim.x * (blockDim.x >> 5);

  for (int row = wid; row < rows; row += nw) {
    const size_t base = (size_t)row * ROWLEN;
    const float* __restrict__ lrow = l + base;
    const float* __restrict__ urow = u + base;

    // Prefetch next row this wave will process (global_prefetch_b8).
    if (row + nw < rows) {
      const size_t nbase = (size_t)(row + nw) * ROWLEN;
      __builtin_prefetch(l + nbase + (size_t)lane * 32, 0, 3);
      __builtin_prefetch(u + nbase + (size_t)lane * 32, 0, 3);
    }

    // ---- load row into registers (coalesced b32: row stride 4100B is only
    // 4B-aligned, so dwordx4 is illegal) ----
    float lw[EPL], uw[EPL];
#pragma unroll
    for (int k = 0; k < EPL; ++k) {
      const int e = k * 32 + lane;
      lw[k] = lrow[e];
      uw[k] = urow[e];
    }

    // ---- 4 fused reductions: lo/hi of l and of u ----
    float lo_l = 0.f, hi_l = 0.f, lo_u = 0.f, hi_u = 0.f;
#pragma unroll
    for (int k = 0; k < EPL; ++k) {
      const int e = k * 32 + lane;
      const float vl = s_il[e];
      const float vu = s_iu[e];
      float w = lw[k];
      float wp = fmaxf(w, 0.f), wn = fminf(w, 0.f);
      lo_l = fmaf(wp, vl, lo_l); lo_l = fmaf(wn, vu, lo_l);
      hi_l = fmaf(wp, vu, hi_l); hi_l = fmaf(wn, vl, hi_l);
      w = uw[k];
      wp = fmaxf(w, 0.f); wn = fminf(w, 0.f);
      lo_u = fmaf(wp, vl, lo_u); lo_u = fmaf(wn, vu, lo_u);
      hi_u = fmaf(wp, vu, hi_u); hi_u = fmaf(wn, vl, hi_u);
    }

    // ---- wave32 cross-lane reduce (butterfly) ----
#pragma unroll
    for (int off = 16; off >= 1; off >>= 1) {
      lo_l += __shfl_xor(lo_l, off, 32);
      hi_l += __shfl_xor(hi_l, off, 32);
      lo_u += __shfl_xor(lo_u, off, 32);
      hi_u += __shfl_xor(hi_u, off, 32);
    }

    const float b_l = lrow[DD];
    const float b_u = urow[DD];
    const float conc_lb = lo_l + b_l;
    const float max_lb  = hi_l + b_l;
    const float min_ub  = lo_u + b_u;
    const float conc_ub = hi_u + b_u;

    // ---- per-row ReLU relaxation scalars (uniform across lanes) ----
    const bool inactive = (conc_ub <= 0.f);
    const bool unstable = (conc_lb < 0.f) & (conc_ub > 0.f);
    const bool mostly_inactive =
        unstable & ((fabsf(conc_lb) > fabsf(conc_ub)) | (max_lb <= 0.f));
    const bool mostly_active = unstable & (fabsf(conc_lb) <= fabsf(conc_ub));

    float s_l;
    if (mostly_active) {
      s_l = (max_lb < 0.f) ? 0.f : max_lb / (max_lb - conc_lb);
    } else {
      s_l = (inactive | mostly_inactive) ? 0.f : 1.f;
    }

    const bool zero_cross = unstable & (min_ub <= 0.f);
    float s_u, delta;
    if (zero_cross) {
      const float a_u = conc_ub / (conc_ub - min_ub);
      s_u = a_u;
      delta = -a_u * min_ub;
    } else {
      s_u = inactive ? 0.f : 1.f;
      delta = 0.f;
    }

    // ---- scaled writeback; fused post-concretization (scales >= 0) ----
    float* __restrict__ pl = out_l + base;
    float* __restrict__ pu = out_u + base;
#pragma unroll
    for (int k = 0; k < EPL; ++k) {
      const int e = k * 32 + lane;
      pl[e] = s_l * lw[k];
      pu[e] = s_u * uw[k];
    }
    if (lane == 0) {
      pl[DD] = s_l * b_l;
      pu[DD] = s_u * b_u + delta;
      out_clb[row] = fmaxf(s_l * conc_lb, 0.f);
      out_cub[row] = fmaxf(s_u * conc_ub + delta, 0.f);
    }
  }
}

extern "C" void kernel_launch(void* const* d_in, const int* in_sizes, int n_in,
                              void* d_out, int out_size, void* d_ws,
                              size_t ws_size, hipStream_t stream) {
  const float* l  = (const float*)d_in[0];
  const float* u  = (const float*)d_in[1];
  const float* il = (const float*)d_in[2];
  const float* iu = (const float*)d_in[3];

  const int rows = in_sizes[0] / ROWLEN;  // 8*4096 = 32768

  float* out_l   = (float*)d_out;
  float* out_u   = out_l + (size_t)rows * ROWLEN;
  float* out_clb = out_u + (size_t)rows * ROWLEN;
  float* out_cub = out_clb + rows;

  const int blocks = 1024;  // 8192 waves, grid-stride over 32768 rows
  relu_deeppoly_fused<<<blocks, 256, 0, stream>>>(l, u, il, iu, out_l, out_u,
                                                  out_clb, out_cub, rows);
}